// Model_11536282157284
// MI455X (gfx1250) — compile-verified
//
#include <hip/hip_runtime.h>
#include <math.h>

// ---------------- problem constants (match reference) ----------------
#define N_NODES 4096
#define N_EDGES 256
#define E_INC   16384
#define BATCH   16
#define D_IN    128
#define D_OUT   128
#define GAMMA_C 4.2f
#define NEG_SLOPE 0.2f
#define BD      (BATCH * D_OUT)        // 2048 floats per [b,d] slab

typedef __attribute__((ext_vector_type(2))) float v2f;
typedef __attribute__((ext_vector_type(8))) float v8f;

// ---------------- workspace layout (float offsets) ----------------
#define WS_X1        0L                               // [N][B][D]  8388608
#define WS_ESUM      (WS_X1 + (long)N_NODES*BD)       // [M][B][D]   524288
#define WS_OUTE      (WS_ESUM + (long)N_EDGES*BD)     // [M][B][D]   524288
#define WS_PNODE     (WS_OUTE + (long)N_EDGES*BD)     // [N][B]       65536
#define WS_QATT      (WS_PNODE + (long)N_NODES*BATCH) // [M][B]        4096
#define WS_NRM2      (WS_QATT + (long)N_EDGES*BATCH)  // [M][B]        4096
#define WS_ABUF      (WS_NRM2 + (long)N_EDGES*BATCH)  // [E][B]      262144
#define WS_ALPHA     (WS_ABUF + (long)E_INC*BATCH)    // [E][B]      262144
#define WS_SEGMAX    (WS_ALPHA + (long)E_INC*BATCH)   // [N][B]       65536
#define WS_DENOM     (WS_SEGMAX + (long)N_NODES*BATCH)// [N][B]       65536
#define WS_DDEG      (WS_DENOM + (long)N_NODES*BATCH) // [N]           4096
#define WS_EDEG      (WS_DDEG + (long)N_NODES)        // [M]            256
#define WS_NROW      (WS_EDEG + (long)N_EDGES)        // [N]           4096
#define WS_EROW      (WS_NROW + (long)N_NODES)        // [M]            256
#define WS_SCAL      (WS_EROW + (long)N_EDGES)        // diff_acc, loss_acc

// ---------------- helpers ----------------
__global__ void fill_f32(float* p, long n, float v) {
    long i = (long)blockIdx.x * blockDim.x + threadIdx.x;
    long s = (long)gridDim.x * blockDim.x;
    for (; i < n; i += s) p[i] = v;
}

__device__ __forceinline__ void atomicMaxFloat(float* addr, float value) {
    if (value >= 0.0f) atomicMax((int*)addr, __float_as_int(value));
    else               atomicMin((unsigned int*)addr, __float_as_uint(value));
}

__device__ __forceinline__ float blockReduceAdd(float v) {
    __shared__ float red[256];
    red[threadIdx.x] = v;
    __syncthreads();
    for (int off = blockDim.x >> 1; off > 0; off >>= 1) {
        if ((int)threadIdx.x < off) red[threadIdx.x] += red[threadIdx.x + off];
        __syncthreads();
    }
    return red[0];
}

// ---------------- K1: WMMA GEMM  x1[n][b][o] = sum_i x[b][n][i] * W[i][o] ---
// One wave per node n: D-tile rows = batches (16), cols = D_OUT in 8 tiles of 16.
// A operand: per-lane 8B contiguous load of x; B operand staged in LDS (full W).
__global__ void __launch_bounds__(128)
k_gemm_wmma(const float* __restrict__ x, const float* __restrict__ W,
            float* __restrict__ x1) {
    __shared__ float Wlds[D_IN * D_OUT];   // 64 KB of the 320 KB WGP LDS
    // cooperative load of W (16384 floats, 128 threads -> 32 float4 each)
    {
        const float4* Wv = (const float4*)W;
        float4* Lv = (float4*)Wlds;
        for (int i = threadIdx.x; i < (D_IN * D_OUT) / 4; i += blockDim.x)
            Lv[i] = Wv[i];
    }
    __syncthreads();

    const int lane  = threadIdx.x & 31;
    const int wave  = threadIdx.x >> 5;
    const int n     = blockIdx.x * 4 + wave;   // 1024 blocks x 4 waves = 4096 nodes
    const int m16   = lane & 15;               // row-in-tile == batch b
    const int khalf = lane >> 4;               // K-half select per ISA A layout

    const float* xrow = x + ((size_t)m16 * N_NODES + n) * D_IN;   // x[b][n][:]

    v8f acc[8] = {};
    for (int k0 = 0; k0 < D_IN; k0 += 4) {
        const int ka = k0 + khalf * 2;
        const v2f av = *(const v2f*)(xrow + ka);          // A[m16][ka..ka+1]
#pragma unroll
        for (int t = 0; t < 8; ++t) {
            const int o = t * 16 + m16;
            v2f bv;
            bv.x = Wlds[ka * D_OUT + o];                  // B[2*khalf  ][col]
            bv.y = Wlds[(ka + 1) * D_OUT + o];            // B[2*khalf+1][col]
            acc[t] = __builtin_amdgcn_wmma_f32_16x16x4_f32(
                false, av, false, bv, (short)0, acc[t], false, false);
        }
    }
    // D layout: VGPR v, lanes 0-15 -> row v, lanes 16-31 -> row v+8; col = m16
    float* dst = x1 + (size_t)n * BD;
#pragma unroll
    for (int t = 0; t < 8; ++t)
#pragma unroll
        for (int v = 0; v < 8; ++v)
            dst[(size_t)(v + 8 * khalf) * D_OUT + t * 16 + m16] = acc[t][v];
}

// ---------------- K2: edge_sums[m] = sum over members (edge_idx sorted) -----
__global__ void __launch_bounds__(256)
k_edge_sums(const float* __restrict__ x1, const int* __restrict__ node_idx,
            const int* __restrict__ edge_idx, float* __restrict__ esum,
            float* __restrict__ Edeg) {
    const int m = blockIdx.x;
    int lo = 0, hi = E_INC;
    while (lo < hi) { int mid = (lo + hi) >> 1; if (edge_idx[mid] <  m) lo = mid + 1; else hi = mid; }
    const int beg = lo;
    lo = beg; hi = E_INC;
    while (lo < hi) { int mid = (lo + hi) >> 1; if (edge_idx[mid] <= m) lo = mid + 1; else hi = mid; }
    const int end = lo;

    const int base = threadIdx.x * 8;          // 256 threads x 8 floats = 2048
    float acc[8] = {};
    for (int e = beg; e < end; ++e) {
        const float* src = x1 + (size_t)node_idx[e] * BD + base;
#pragma unroll
        for (int j = 0; j < 8; ++j) acc[j] += src[j];
    }
    float* dst = esum + (size_t)m * BD + base;
#pragma unroll
    for (int j = 0; j < 8; ++j) dst[j] = acc[j];
    if (threadIdx.x == 0) Edeg[m] = (float)(end - beg);
}

// ---------------- K3: per-edge att dot, squared norm, rowsum ----------------
__global__ void k_edge_stats(const float* __restrict__ esum, const float* __restrict__ att,
                             float* __restrict__ qatt, float* __restrict__ nrm2,
                             float* __restrict__ erow) {
    const int i = blockIdx.x * blockDim.x + threadIdx.x;   // M*B = 4096
    if (i >= N_EDGES * BATCH) return;
    const int m = i >> 4, b = i & 15;
    const float* q = esum + (size_t)m * BD + b * D_OUT;
    float qa = 0.f, n2 = 0.f, s = 0.f;
    for (int d = 0; d < D_OUT; ++d) {
        float v = q[d];
        qa += v * att[D_OUT + d];
        n2 += v * v;
        s  += v;
    }
    qatt[i] = qa; nrm2[i] = n2;
    atomicAdd(&erow[m], s);
}

// ---------------- K4: O(M^2) pairwise contrastive loss ----------------------
__global__ void __launch_bounds__(256)
k_loss_hyper(const float* __restrict__ esum, const float* __restrict__ nrm2,
             float* __restrict__ loss_acc) {
    const int p = blockIdx.x * blockDim.x + threadIdx.x;   // M*M = 65536
    float meanv = 0.f;
    if (p < N_EDGES * N_EDGES) {
        const int k = p >> 8, m = p & 255;
        const float* qk = esum + (size_t)k * BD;
        const float* qm = esum + (size_t)m * BD;
        for (int b = 0; b < BATCH; ++b) {
            float inner = 0.f;
            const float* a = qk + b * D_OUT;
            const float* c = qm + b * D_OUT;
            for (int d = 0; d < D_OUT; ++d) inner += a[d] * c[d];
            const float n2k = nrm2[k * BATCH + b], n2m = nrm2[m * BATCH + b];
            const float ac = inner / (sqrtf(n2k) * sqrtf(n2m));
            const float d2 = fmaxf(n2k + n2m - 2.f * inner, 0.f);
            const float dist = sqrtf(d2 + 1e-12f);
            meanv += ac * dist + (1.f - ac) * fmaxf(GAMMA_C - dist, 0.f);
        }
        meanv = fabsf(meanv / (float)BATCH);
    }
    float tot = blockReduceAdd(meanv);
    if (threadIdx.x == 0) atomicAdd(loss_acc, tot);
}

// ---------------- K5: per-node att dot + rowsum -----------------------------
__global__ void k_node_stats(const float* __restrict__ x1, const float* __restrict__ att,
                             float* __restrict__ pnode, float* __restrict__ nrow) {
    const int i = blockIdx.x * blockDim.x + threadIdx.x;   // N*B = 65536
    if (i >= N_NODES * BATCH) return;
    const int n = i >> 4, b = i & 15;
    const float* q = x1 + (size_t)n * BD + b * D_OUT;
    float pa = 0.f, s = 0.f;
    for (int d = 0; d < D_IN; ++d) { float v = q[d]; pa += v * att[d]; s += v; }
    pnode[i] = pa;
    atomicAdd(&nrow[n], s);
}

// ---------------- K5b: node degrees ----------------------------------------
__global__ void k_node_deg(const int* __restrict__ node_idx, float* __restrict__ Ddeg) {
    const int e = blockIdx.x * blockDim.x + threadIdx.x;
    if (e < E_INC) atomicAdd(&Ddeg[node_idx[e]], 1.0f);
}

// ---------------- K6: leaky-relu logits + segment max -----------------------
__global__ void k_logits(const int* __restrict__ node_idx, const int* __restrict__ edge_idx,
                         const float* __restrict__ pnode, const float* __restrict__ qatt,
                         float* __restrict__ abuf, float* __restrict__ segmax) {
    const int i = blockIdx.x * blockDim.x + threadIdx.x;   // E*B = 262144
    if (i >= E_INC * BATCH) return;
    const int e = i >> 4, b = i & 15;
    float v = pnode[node_idx[e] * BATCH + b] + qatt[edge_idx[e] * BATCH + b];
    v = (v >= 0.f) ? v : NEG_SLOPE * v;
    abuf[i] = v;
    atomicMaxFloat(&segmax[node_idx[e] * BATCH + b], v);
}

// ---------------- K7: exp + denom -------------------------------------------
__global__ void k_exp_denom(const int* __restrict__ node_idx,
                            float* __restrict__ abuf, const float* __restrict__ segmax,
                            float* __restrict__ denom) {
    const int i = blockIdx.x * blockDim.x + threadIdx.x;
    if (i >= E_INC * BATCH) return;
    const int e = i >> 4, b = i & 15;
    const float v = __expf(abuf[i] - segmax[node_idx[e] * BATCH + b]);
    abuf[i] = v;
    atomicAdd(&denom[node_idx[e] * BATCH + b], v);
}

// ---------------- K8: normalize alpha ---------------------------------------
__global__ void k_alpha(const int* __restrict__ node_idx, const float* __restrict__ abuf,
                        const float* __restrict__ denom, float* __restrict__ alpha) {
    const int i = blockIdx.x * blockDim.x + threadIdx.x;
    if (i >= E_INC * BATCH) return;
    const int e = i >> 4, b = i & 15;
    alpha[i] = abuf[i] / denom[node_idx[e] * BATCH + b];
}

// ---------------- Kd: sum over incidence of (x_i - x_j) ---------------------
__global__ void __launch_bounds__(256)
k_diff(const int* __restrict__ node_idx, const int* __restrict__ edge_idx,
       const float* __restrict__ nrow, const float* __restrict__ erow,
       float* __restrict__ diff_acc) {
    const int e = blockIdx.x * blockDim.x + threadIdx.x;
    float v = 0.f;
    if (e < E_INC) v = nrow[node_idx[e]] - erow[edge_idx[e]];
    float tot = blockReduceAdd(v);
    if (threadIdx.x == 0) atomicAdd(diff_acc, tot);
}

// ---------------- K9: propagate node -> hyperedge ---------------------------
__global__ void __launch_bounds__(256)
k_prop1(const float* __restrict__ x1, const int* __restrict__ node_idx,
        const int* __restrict__ edge_idx, const float* __restrict__ alpha,
        const float* __restrict__ Edeg, float* __restrict__ out_e) {
    const int m = blockIdx.x;
    int lo = 0, hi = E_INC;
    while (lo < hi) { int mid = (lo + hi) >> 1; if (edge_idx[mid] <  m) lo = mid + 1; else hi = mid; }
    const int beg = lo;
    lo = beg; hi = E_INC;
    while (lo < hi) { int mid = (lo + hi) >> 1; if (edge_idx[mid] <= m) lo = mid + 1; else hi = mid; }
    const int end = lo;

    const float ed = Edeg[m];
    const float bn = (ed > 0.f) ? (1.f / fmaxf(ed, 1.f)) : 0.f;
    const int base = threadIdx.x * 8;
    const int b = base >> 7;                    // 8-float chunk lies in one batch
    float acc[8] = {};
    for (int e = beg; e < end; ++e) {
        const float al = alpha[e * BATCH + b];
        const float* src = x1 + (size_t)node_idx[e] * BD + base;
#pragma unroll
        for (int j = 0; j < 8; ++j) acc[j] += al * src[j];
    }
    float* dst = out_e + (size_t)m * BD + base;
#pragma unroll
    for (int j = 0; j < 8; ++j) dst[j] = bn * acc[j];
}

// ---------------- K10: propagate hyperedge -> node (scatter, atomic) --------
__global__ void k_prop2(const float* __restrict__ out_e, const int* __restrict__ node_idx,
                        const int* __restrict__ edge_idx, const float* __restrict__ alpha,
                        const float* __restrict__ Ddeg, float* __restrict__ out) {
    const long id = (long)blockIdx.x * blockDim.x + threadIdx.x;  // E*256
    if (id >= (long)E_INC * 256) return;
    const int e = (int)(id >> 8);
    const int base = ((int)id & 255) * 8;       // position in [b*128+d] slab
    const int b = base >> 7;
    const int d0 = base & (D_OUT - 1);
    const int n = node_idx[e], m = edge_idx[e];
    const float scale = Ddeg[n] * alpha[e * BATCH + b];
    const float* src = out_e + (size_t)m * BD + base;
    float* dst = out + ((size_t)b * N_NODES + n) * D_OUT + d0;    // out[B][N][D]
#pragma unroll
    for (int j = 0; j < 8; ++j) atomicAdd(dst + j, scale * src[j]);
}

// ---------------- K11: final scalar -----------------------------------------
__global__ void k_final(const float* __restrict__ scal, float* __restrict__ out) {
    const float diff_mean = scal[0] / ((float)E_INC * BATCH * D_OUT);
    const float loss = scal[1] / ((float)(N_EDGES + 1) * (float)(N_EDGES + 1));
    out[(size_t)BATCH * N_NODES * D_OUT] = fabsf(diff_mean) + loss;
}

// ============================================================================
extern "C" void kernel_launch(void* const* d_in, const int* in_sizes, int n_in,
                              void* d_out, int out_size, void* d_ws, size_t ws_size,
                              hipStream_t stream) {
    const float* x    = (const float*)d_in[0];              // [B,N,D_IN]
    const int*   hidx = (const int*)d_in[1];                // [2,E]
    const float* W    = (const float*)d_in[2];              // [D_IN,D_OUT]
    const float* att  = (const float*)d_in[3];              // [2*D_OUT]
    const int* node_idx = hidx;
    const int* edge_idx = hidx + E_INC;

    float* ws   = (float*)d_ws;
    float* x1   = ws + WS_X1;
    float* esum = ws + WS_ESUM;
    float* oute = ws + WS_OUTE;
    float* pnod = ws + WS_PNODE;
    float* qatt = ws + WS_QATT;
    float* nrm2 = ws + WS_NRM2;
    float* abuf = ws + WS_ABUF;
    float* alfa = ws + WS_ALPHA;
    float* smax = ws + WS_SEGMAX;
    float* deno = ws + WS_DENOM;
    float* ddeg = ws + WS_DDEG;
    float* edeg = ws + WS_EDEG;
    float* nrow = ws + WS_NROW;
    float* erow = ws + WS_EROW;
    float* scal = ws + WS_SCAL;
    float* out  = (float*)d_out;

    // init: output zero (atomic scatter target), atomic accumulators zero,
    // segment-max to -inf
    const long zero_small = (WS_SCAL + 2) - WS_DENOM;       // denom..scalars
    fill_f32<<<1024, 256, 0, stream>>>(out, (long)out_size, 0.0f);
    fill_f32<<<256, 256, 0, stream>>>(ws + WS_DENOM, zero_small, 0.0f);
    fill_f32<<<256, 256, 0, stream>>>(smax, (long)N_NODES * BATCH,
                                      -__builtin_huge_valf());

    // 1) x1 = x @ W  (WMMA f32 16x16x4; 1 node per wave, W staged in LDS)
    k_gemm_wmma<<<N_NODES / 4, 128, 0, stream>>>(x, W, x1);

    // 2) per-hyperedge member sums (+ edge degrees)
    k_edge_sums<<<N_EDGES, 256, 0, stream>>>(x1, node_idx, edge_idx, esum, edeg);

    // 3) per-edge att dot / norms / rowsums
    k_edge_stats<<<(N_EDGES * BATCH) / 256, 256, 0, stream>>>(esum, att, qatt, nrm2, erow);

    // 4) O(M^2) contrastive loss
    k_loss_hyper<<<(N_EDGES * N_EDGES) / 256, 256, 0, stream>>>(esum, nrm2, scal + 1);

    // 5) per-node att dot / rowsums ; node degrees
    k_node_stats<<<(N_NODES * BATCH) / 256, 256, 0, stream>>>(x1, att, pnod, nrow);
    k_node_deg<<<E_INC / 256, 256, 0, stream>>>(node_idx, ddeg);

    // 6-8) segment softmax over each node's incident entries
    k_logits<<<(E_INC * BATCH) / 256, 256, 0, stream>>>(node_idx, edge_idx, pnod, qatt, abuf, smax);
    k_exp_denom<<<(E_INC * BATCH) / 256, 256, 0, stream>>>(node_idx, abuf, smax, deno);
    k_alpha<<<(E_INC * BATCH) / 256, 256, 0, stream>>>(node_idx, abuf, deno, alfa);

    // constrain term: mean(x_i - x_j) via precomputed rowsums
    k_diff<<<E_INC / 256, 256, 0, stream>>>(node_idx, edge_idx, nrow, erow, scal);

    // 9) node -> hyperedge ; 10) hyperedge -> node (into [B,N,D] output)
    k_prop1<<<N_EDGES, 256, 0, stream>>>(x1, node_idx, edge_idx, alfa, edeg, oute);
    k_prop2<<<(E_INC * 256) / 256, 256, 0, stream>>>(oute, node_idx, edge_idx, alfa, ddeg, out);

    // 11) scalar tail
    k_final<<<1, 1, 0, stream>>>(scal, out);
}